// ManifoldNormGr_50766513438853
// MI455X (gfx1250) — compile-verified
//
#include <hip/hip_runtime.h>
#include <hip/hip_bf16.h>
#include <math.h>

// ---------------------------------------------------------------------------
// ManifoldNormGr on MI455X (gfx1250): B=256, C=64, N=64, P=16, fp32 all the way.
// Every matmul runs on V_WMMA_F32_16X16X4_F32 (fp32 matrix pipe, wave32).
// 4KB tiles staged into LDS via the Tensor Data Mover (tensor_load_to_lds,
// TENSORcnt). One wave (32 threads) per block => EXEC all-ones at every WMMA.
// ---------------------------------------------------------------------------

#define BB 256
#define CC 64
#define NN 64
#define PP 16

// ws layout (floats)
#define WS_PARTIAL 0
#define WS_MEAN (CC * 16 * NN * PP)            // 64*16*1024 = 1048576
#define WS_LFS  (WS_MEAN + CC * NN * PP)       // + 64*1024  = 1114112
// total: WS_LFS + CC*NN*NN = 1376256 floats = 5.25 MB

typedef float v2f __attribute__((ext_vector_type(2)));
typedef float v4f __attribute__((ext_vector_type(4)));
typedef float v8f __attribute__((ext_vector_type(8)));
typedef unsigned int v4u __attribute__((ext_vector_type(4)));
typedef int v4i __attribute__((ext_vector_type(4)));
typedef int v8i __attribute__((ext_vector_type(8)));

// ----------------------- Tensor Data Mover: 4KB contiguous tile -> LDS ------
// D# per CDNA5 ISA: group0 = {flags, lds_addr, gaddr_lo,
// gaddr_hi|type=2}; group1 packs data_size=4B, tensor_dim0=tile_dim0=1024,
// tensor_dim1=tile_dim1=1, tensor_dim0_stride=1024. Groups 2/3/extra unused
// (<=2D tensor). Toolchain exposes the 6-arg builtin (g0,g1,g2,g3,g4,cpol).
__device__ __forceinline__ void tdm_load_4kb(void* lds_ptr, const void* gptr) {
  unsigned long long ga = (unsigned long long)(uintptr_t)gptr;
  unsigned int ldsoff = (unsigned int)(uintptr_t)lds_ptr;  // low 32 bits = LDS byte offset
  v4u g0;
  g0[0] = 1u;                                             // count=1, user mode
  g0[1] = ldsoff;                                         // lds_addr (bits 63:32)
  g0[2] = (unsigned int)ga;                               // global_addr[31:0]
  g0[3] = (unsigned int)((ga >> 32) & 0x01FFFFFFu) | (2u << 30);  // addr[56:32] | type=2
  v8i g1;
  g1[0] = 0x00020000;          // workgroup_mask=0, data_size=2 (4 bytes)
  g1[1] = (int)(1024u << 16);  // tensor_dim0[15:0] -> bits[63:48]  (dim0=1024)
  g1[2] = 0x00010000;          // tensor_dim0[31:16]=0, tensor_dim1 lo16 = 1
  g1[3] = (int)(1024u << 16);  // tensor_dim1 hi16 = 0, tile_dim0 = 1024
  g1[4] = 1;                   // tile_dim1 = 1, tile_dim2 = 0
  g1[5] = 1024;                // tensor_dim0_stride[31:0]
  g1[6] = 0;                   // tensor_dim0_stride[47:32], tensor_dim1_stride lo
  g1[7] = 0;
  v4i z4 = {0, 0, 0, 0};
  v8i z8 = {0, 0, 0, 0, 0, 0, 0, 0};
  __builtin_amdgcn_tensor_load_to_lds(g0, g1, z4, z4, z8, 0);
}
__device__ __forceinline__ void tdm_wait() {
  __builtin_amdgcn_s_wait_tensorcnt(0);
}

// ----------------------- WMMA fragment helpers (f32, 16x16x4) ---------------
// A (16x4, f32): VGPR0: lanes0-15 K=0 | lanes16-31 K=2 ; VGPR1: K=1 | K=3
__device__ __forceinline__ v2f ldA(const float* A, int lda, int mr, int kb, int lane) {
  int half = lane >> 4, m = lane & 15;
  return *(const v2f*)(A + (mr + m) * lda + kb + 2 * half);  // contiguous pair
}
// Fragment of A^T where A is stored row-major (lda = stride of stored A)
__device__ __forceinline__ v2f ldAT(const float* A, int lda, int mr, int kb, int lane) {
  int half = lane >> 4, m = lane & 15;
  v2f a;
  a.x = A[(kb + 2 * half) * lda + mr + m];
  a.y = A[(kb + 2 * half + 1) * lda + mr + m];
  return a;
}
// B (4x16, f32): mirrored split: VGPR0: K=0 | K=2 ; VGPR1: K=1 | K=3
__device__ __forceinline__ v2f ldB(const float* Bm, int ldb, int kb, int nc, int lane) {
  int half = lane >> 4, n = lane & 15;
  v2f b;
  b.x = Bm[(kb + 2 * half) * ldb + nc + n];
  b.y = Bm[(kb + 2 * half + 1) * ldb + nc + n];
  return b;
}
// Fragment of B^T where B is stored row-major
__device__ __forceinline__ v2f ldBT(const float* Bm, int ldb, int kb, int nc, int lane) {
  int half = lane >> 4, n = lane & 15;
  return *(const v2f*)(Bm + (nc + n) * ldb + kb + 2 * half);  // contiguous pair
}
// C/D (16x16 f32): VGPR r: lanes0-15 -> M=r, lanes16-31 -> M=r+8; N=lane&15
__device__ __forceinline__ void stC(float* Cm, int ldc, int mr, int nc, int lane, v8f c) {
  int half = lane >> 4, n = lane & 15;
#pragma unroll
  for (int r = 0; r < 8; ++r) Cm[(size_t)(mr + r + 8 * half) * ldc + nc + n] = c[r];
}

__device__ __forceinline__ v8f wmma4(v2f a, v2f b, v8f c) {
  return __builtin_amdgcn_wmma_f32_16x16x4_f32(false, a, false, b, (short)0, c,
                                               false, false);
}

__device__ __forceinline__ v8f gemm_nn(const float* A, int lda, const float* Bm, int ldb,
                                       int mr, int nc, int K, v8f acc, int lane) {
#pragma unroll
  for (int kb = 0; kb < K; kb += 4)
    acc = wmma4(ldA(A, lda, mr, kb, lane), ldB(Bm, ldb, kb, nc, lane), acc);
  return acc;
}
__device__ __forceinline__ v8f gemm_tn(const float* A, int lda, const float* Bm, int ldb,
                                       int mr, int nc, int K, v8f acc, int lane) {
#pragma unroll
  for (int kb = 0; kb < K; kb += 4)
    acc = wmma4(ldAT(A, lda, mr, kb, lane), ldB(Bm, ldb, kb, nc, lane), acc);
  return acc;
}
__device__ __forceinline__ v8f gemm_nt(const float* A, int lda, const float* Bm, int ldb,
                                       int mr, int nc, int K, v8f acc, int lane) {
#pragma unroll
  for (int kb = 0; kb < K; kb += 4)
    acc = wmma4(ldA(A, lda, mr, kb, lane), ldBT(Bm, ldb, kb, nc, lane), acc);
  return acc;
}

__device__ __forceinline__ v8f vzero() {
  v8f z = {0.f, 0.f, 0.f, 0.f, 0.f, 0.f, 0.f, 0.f};
  return z;
}

// ----------------------- 16x16 symmetric Jacobi eigensolver ------------------
// G (16x16 LDS) -> diag(G)=eigvals, V (16x16 LDS)=eigvecs. 8 disjoint rotation
// pairs per round (round-robin), applied wave-parallel.
__device__ __forceinline__ void jacobi16(float* G, float* V, float* rc, float* rs,
                                         int* pq, int lane) {
  for (int i = lane; i < 256; i += 32) V[i] = ((i >> 4) == (i & 15)) ? 1.f : 0.f;
  __syncthreads();
  for (int sweep = 0; sweep < 6; ++sweep) {
    for (int r = 0; r < 15; ++r) {
      if (lane < 8) {
        int p, q;
        if (lane == 0) { p = 15; q = r; }
        else { p = (r + lane) % 15; q = (r + 15 - lane) % 15; }
        float app = G[p * 16 + p], aqq = G[q * 16 + q], apq = G[p * 16 + q];
        float cth = 1.f, sth = 0.f;
        if (fabsf(apq) > 1e-12f) {
          float tau = (aqq - app) / (2.f * apq);
          float t = (tau >= 0.f ? 1.f : -1.f) / (fabsf(tau) + sqrtf(1.f + tau * tau));
          cth = rsqrtf(1.f + t * t);
          sth = t * cth;
        }
        rc[lane] = cth; rs[lane] = sth; pq[lane] = p; pq[8 + lane] = q;
      }
      __syncthreads();
      // column update: G <- G*J ; V <- V*J  (128 independent 2-col updates)
      for (int t = lane; t < 128; t += 32) {
        int row = t & 15, k = t >> 4;
        int p = pq[k], q = pq[8 + k];
        float cth = rc[k], sth = rs[k];
        float gp = G[row * 16 + p], gq = G[row * 16 + q];
        G[row * 16 + p] = cth * gp - sth * gq;
        G[row * 16 + q] = sth * gp + cth * gq;
        float vp = V[row * 16 + p], vq = V[row * 16 + q];
        V[row * 16 + p] = cth * vp - sth * vq;
        V[row * 16 + q] = sth * vp + cth * vq;
      }
      __syncthreads();
      // row update: G <- J^T * G
      for (int t = lane; t < 128; t += 32) {
        int col = t & 15, k = t >> 4;
        int p = pq[k], q = pq[8 + k];
        float cth = rc[k], sth = rs[k];
        float gp = G[p * 16 + col], gq = G[q * 16 + col];
        G[p * 16 + col] = cth * gp - sth * gq;
        G[q * 16 + col] = sth * gp + cth * gq;
      }
      __syncthreads();
    }
  }
}

// ----------------------- 16x16 inverse via Newton-Schulz (WMMA) --------------
__device__ __forceinline__ void inv16(const float* M, float* Y, float* Ta, float* Tb,
                                      int lane) {
  if (lane < 16) {
    float rsum = 0.f, csum = 0.f;
    for (int j = 0; j < 16; ++j) {
      rsum += fabsf(M[lane * 16 + j]);
      csum += fabsf(M[j * 16 + lane]);
    }
    Ta[lane] = rsum; Ta[16 + lane] = csum;
  }
  __syncthreads();
  if (lane == 0) {
    float mr = 0.f, mc = 0.f;
    for (int j = 0; j < 16; ++j) { mr = fmaxf(mr, Ta[j]); mc = fmaxf(mc, Ta[16 + j]); }
    Tb[0] = 1.f / fmaxf(mr * mc, 1e-20f);
  }
  __syncthreads();
  float alpha = Tb[0];
  __syncthreads();
  for (int i = lane; i < 256; i += 32) Y[i] = alpha * M[(i & 15) * 16 + (i >> 4)];
  __syncthreads();
  for (int it = 0; it < 12; ++it) {
    v8f acc = gemm_nn(M, 16, Y, 16, 0, 0, 16, vzero(), lane);   // Ta = M*Y
    stC(Ta, 16, 0, 0, lane, acc);
    __syncthreads();
    for (int i = lane; i < 256; i += 32)
      Ta[i] = (((i >> 4) == (i & 15)) ? 2.f : 0.f) - Ta[i];     // 2I - M*Y
    __syncthreads();
    v8f acc2 = gemm_nn(Y, 16, Ta, 16, 0, 0, 16, vzero(), lane); // Tb = Y*(2I-MY)
    stC(Tb, 16, 0, 0, lane, acc2);
    __syncthreads();
    for (int i = lane; i < 256; i += 32) Y[i] = Tb[i];
    __syncthreads();
  }
}

// W = V * diag(fd) * V^T   (Wtmp scratch 256)
__device__ __forceinline__ void vfvT(const float* V, const float* fd, float* Wtmp,
                                     float* Wout, int lane) {
  for (int i = lane; i < 256; i += 32) Wtmp[i] = V[i] * fd[i & 15];
  __syncthreads();
  v8f acc = gemm_nt(Wtmp, 16, V, 16, 0, 0, 16, vzero(), lane);
  stC(Wout, 16, 0, 0, lane, acc);
  __syncthreads();
}

// Grassmann log map: Dout = logmap(Ms, Xs), both 64x16 in LDS.
__device__ __forceinline__ void logmap_dev(const float* Ms, const float* Xs, float* Ts,
                                           float* Dout, float* M16, float* Pinv,
                                           float* Ga, float* Va, float* W1, float* W2,
                                           float* rc, float* rs, int* pq, float* fd,
                                           int lane) {
  // M16 = Ms^T * Xs   (16x16, K=64)
  v8f acc = gemm_tn(Ms, 16, Xs, 16, 0, 0, 64, vzero(), lane);
  stC(M16, 16, 0, 0, lane, acc);
  __syncthreads();
  inv16(M16, Pinv, W1, W2, lane);
  // Ts = Xs*Pinv - Ms  (64x16)
  for (int mr = 0; mr < 64; mr += 16) {
    v8f a = gemm_nn(Xs, 16, Pinv, 16, mr, 0, 16, vzero(), lane);
    stC(Ts, 16, mr, 0, lane, a);
  }
  __syncthreads();
  for (int i = lane; i < 1024; i += 32) Ts[i] -= Ms[i];
  __syncthreads();
  // Ga = Ts^T * Ts
  v8f g = gemm_tn(Ts, 16, Ts, 16, 0, 0, 64, vzero(), lane);
  stC(Ga, 16, 0, 0, lane, g);
  __syncthreads();
  jacobi16(Ga, Va, rc, rs, pq, lane);
  if (lane < 16) {
    float s = sqrtf(fmaxf(Ga[lane * 17], 0.f));
    fd[lane] = (s < 1e-8f) ? 1.f : atanf(s) / s;
  }
  __syncthreads();
  vfvT(Va, fd, W1, W2, lane);                       // W2 = V atanc(S) V^T
  for (int mr = 0; mr < 64; mr += 16) {             // Dout = Ts * W2
    v8f a = gemm_nn(Ts, 16, W2, 16, mr, 0, 16, vzero(), lane);
    stC(Dout, 16, mr, 0, lane, a);
  }
  __syncthreads();
}

#define SHARED_DECL                                                    \
  __shared__ __align__(16) float SM[6 * 1024 + 8 * 256];               \
  __shared__ float rc[8], rs[8], fd[16], fd2[16], fd3[16];             \
  __shared__ int pq[16];                                               \
  float* Xs = SM;            float* Ms = SM + 1024;                    \
  float* Ts = SM + 2048;     float* Ds = SM + 3072;                    \
  float* Hs = SM + 4096;     float* Us = SM + 5120;                    \
  float* M16 = SM + 6144;    float* Pinv = M16 + 256;                  \
  float* Ga = Pinv + 256;    float* Va = Ga + 256;                     \
  float* W1 = Va + 256;      float* W2 = W1 + 256;                     \
  (void)Xs; (void)Ms; (void)Ts; (void)Ds; (void)Hs; (void)Us;          \
  (void)fd2; (void)fd3;

// ======================= Stage A1: partial tangent sums ======================
// grid = C*16 blocks; block (c,g) handles b in [g*16, g*16+16)
__global__ void __launch_bounds__(32) gr_stageA1(const float* __restrict__ X,
                                                 float* __restrict__ ws) {
  SHARED_DECL;
  const int lane = threadIdx.x;
  const int c = blockIdx.x >> 4, g = blockIdx.x & 15;
  // mean0 = X[0, c] via TDM; tangent accumulator -> 0
  tdm_load_4kb(Ms, X + (size_t)c * 1024);
  for (int i = lane; i < 1024; i += 32) Hs[i] = 0.f;
  tdm_wait();
  __syncthreads();
  for (int bi = 0; bi < 16; ++bi) {
    int b = g * 16 + bi;
    tdm_load_4kb(Xs, X + ((size_t)b * CC + c) * 1024);
    tdm_wait();
    __syncthreads();
    logmap_dev(Ms, Xs, Ts, Ds, M16, Pinv, Ga, Va, W1, W2, rc, rs, pq, fd, lane);
    for (int i = lane; i < 1024; i += 32) Hs[i] += Ds[i];
    __syncthreads();
  }
  const float invB = 1.f / (float)BB;
  for (int i = lane; i < 1024; i += 32)
    ws[WS_PARTIAL + (size_t)(c * 16 + g) * 1024 + i] = Hs[i] * invB;
}

// ======================= Stage A2: mean + lfs per channel ====================
__global__ void __launch_bounds__(32) gr_stageA2(const float* __restrict__ X,
                                                 float* __restrict__ ws) {
  SHARED_DECL;
  const int lane = threadIdx.x;
  const int c = blockIdx.x;
  tdm_load_4kb(Ms, X + (size_t)c * 1024);  // mean0
  // Hs = mean tangent (sum of 16 partials), vectorized
  for (int i = lane * 4; i < 1024; i += 128) {
    v4f s = {0.f, 0.f, 0.f, 0.f};
    for (int g = 0; g < 16; ++g)
      s += *(const v4f*)(ws + WS_PARTIAL + (size_t)(c * 16 + g) * 1024 + i);
    *(v4f*)(Hs + i) = s;
  }
  tdm_wait();
  __syncthreads();
  // ---- mean = expmap(Ms, Hs) ----
  v8f g0 = gemm_tn(Hs, 16, Hs, 16, 0, 0, 64, vzero(), lane);
  stC(Ga, 16, 0, 0, lane, g0);
  __syncthreads();
  jacobi16(Ga, Va, rc, rs, pq, lane);
  if (lane < 16) {
    float s = sqrtf(fmaxf(Ga[lane * 17], 0.f));
    fd[lane] = cosf(s);
    fd2[lane] = (s < 1e-8f) ? 1.f : sinf(s) / s;
  }
  __syncthreads();
  vfvT(Va, fd, W1, M16, lane);    // M16 = V cos V^T
  vfvT(Va, fd2, W1, Pinv, lane);  // Pinv = V sinc V^T
  for (int mr = 0; mr < 64; mr += 16) {
    v8f a = gemm_nn(Ms, 16, M16, 16, mr, 0, 16, vzero(), lane);
    a = gemm_nn(Hs, 16, Pinv, 16, mr, 0, 16, a, lane);
    stC(Us, 16, mr, 0, lane, a);  // Us = mean
  }
  __syncthreads();
  for (int i = lane * 4; i < 1024; i += 128)
    *(v4f*)(ws + WS_MEAN + (size_t)c * 1024 + i) = *(const v4f*)(Us + i);
  // ---- Delta = [Mu - inv(Mu)^T ; -Ml] ----
  inv16(Us, Pinv, W1, W2, lane);  // Pinv = inv(Mu), Mu = top 16x16 of mean
  for (int i = lane; i < 256; i += 32)
    Ds[i] = Us[i] - Pinv[(i & 15) * 16 + (i >> 4)];
  for (int i = 256 + lane; i < 1024; i += 32) Ds[i] = -Us[i];
  __syncthreads();
  // ---- SVD of Delta via Gram ----
  v8f g1 = gemm_tn(Ds, 16, Ds, 16, 0, 0, 64, vzero(), lane);
  stC(Ga, 16, 0, 0, lane, g1);
  __syncthreads();
  jacobi16(Ga, Va, rc, rs, pq, lane);
  if (lane < 16) {
    float s = sqrtf(fmaxf(Ga[lane * 17], 0.f));
    float r = rsqrtf(1.f + s * s);
    fd[lane] = r;                      // cos(atan s)
    fd2[lane] = s * r;                 // sin(atan s)
    fd3[lane] = 1.f / fmaxf(s, 1e-6f); // 1/s
  }
  __syncthreads();
  // Q = Delta * V * diag(1/s)  -> Ts
  for (int i = lane; i < 256; i += 32) W1[i] = Va[i] * fd3[i & 15];
  __syncthreads();
  for (int mr = 0; mr < 64; mr += 16) {
    v8f a = gemm_nn(Ds, 16, W1, 16, mr, 0, 16, vzero(), lane);
    stC(Ts, 16, mr, 0, lane, a);
  }
  __syncthreads();
  // Mq = mean * V  -> Xs
  for (int mr = 0; mr < 64; mr += 16) {
    v8f a = gemm_nn(Us, 16, Va, 16, mr, 0, 16, vzero(), lane);
    stC(Xs, 16, mr, 0, lane, a);
  }
  __syncthreads();
  // W = -Mq*sin + Q*(cos-1)  -> Hs
  for (int i = lane; i < 1024; i += 32)
    Hs[i] = -Xs[i] * fd2[i & 15] + Ts[i] * (fd[i & 15] - 1.f);
  __syncthreads();
  // lfs = W * Q^T + I  -> ws
  float* lfs = ws + WS_LFS + (size_t)c * 4096;
  for (int mr = 0; mr < 64; mr += 16)
    for (int nc = 0; nc < 64; nc += 16) {
      v8f a = gemm_nt(Hs, 16, Ts, 16, mr, nc, 16, vzero(), lane);
      int half = lane >> 4, n = lane & 15;
#pragma unroll
      for (int r = 0; r < 8; ++r)
        if (mr + r + 8 * half == nc + n) a[r] += 1.f;
      stC(lfs, 64, mr, nc, lane, a);
    }
}

// ======================= Stage B: per-(b,c) pipeline =========================
__global__ void __launch_bounds__(32) gr_stageB(const float* __restrict__ X,
                                                const float* __restrict__ bias,
                                                const float* __restrict__ shift,
                                                const float* __restrict__ ws,
                                                float* __restrict__ out) {
  SHARED_DECL;
  const int lane = threadIdx.x;
  const int c = blockIdx.x & 63, b = blockIdx.x >> 6;
  const float* Bc = bias + (size_t)c * 4096;
  const float* lf = ws + WS_LFS + (size_t)c * 4096;
  // stage tiles via TDM while prefetching the operator matrices
  tdm_load_4kb(Ms, ws + WS_MEAN + (size_t)c * 1024);
  tdm_load_4kb(Xs, X + ((size_t)b * CC + c) * 1024);
  __builtin_prefetch(Bc + lane * 64, 0, 0);          // global_prefetch_b8
  __builtin_prefetch(lf + lane * 64, 0, 0);
  tdm_wait();
  __syncthreads();
  // delta = logmap(mean, X)  -> Ds
  logmap_dev(Ms, Xs, Ts, Ds, M16, Pinv, Ga, Va, W1, W2, rc, rs, pq, fd, lane);
  // H = shift * (lfs @ delta)  -> Hs
  const float sh = shift[c];
  for (int mr = 0; mr < 64; mr += 16) {
    v8f a = gemm_nn(lf, 64, Ds, 16, mr, 0, 64, vzero(), lane);
    stC(Hs, 16, mr, 0, lane, a);
  }
  __syncthreads();
  for (int i = lane; i < 1024; i += 32) Hs[i] *= sh;
  __syncthreads();
  // U_norm = expmap(I_np, H) = pad(V cos V^T) + H * (V sinc V^T)  -> Us
  v8f g1 = gemm_tn(Hs, 16, Hs, 16, 0, 0, 64, vzero(), lane);
  stC(Ga, 16, 0, 0, lane, g1);
  __syncthreads();
  jacobi16(Ga, Va, rc, rs, pq, lane);
  if (lane < 16) {
    float s = sqrtf(fmaxf(Ga[lane * 17], 0.f));
    fd[lane] = cosf(s);
    fd2[lane] = (s < 1e-8f) ? 1.f : sinf(s) / s;
  }
  __syncthreads();
  vfvT(Va, fd, W1, M16, lane);    // M16 = V cos V^T
  vfvT(Va, fd2, W1, Pinv, lane);  // Pinv = V sinc V^T
  for (int mr = 0; mr < 64; mr += 16) {
    v8f a = gemm_nn(Hs, 16, Pinv, 16, mr, 0, 16, vzero(), lane);
    stC(Us, 16, mr, 0, lane, a);
  }
  __syncthreads();
  for (int i = lane; i < 256; i += 32) Us[i] += M16[i];
  __syncthreads();
  // out[b,c] = bias[c] @ U_norm
  float* oc = out + ((size_t)b * CC + c) * 1024;
  for (int mr = 0; mr < 64; mr += 16) {
    v8f a = gemm_nn(Bc, 64, Us, 16, mr, 0, 64, vzero(), lane);
    stC(oc, 16, mr, 0, lane, a);
  }
}

// ============================================================================
extern "C" void kernel_launch(void* const* d_in, const int* in_sizes, int n_in,
                              void* d_out, int out_size, void* d_ws, size_t ws_size,
                              hipStream_t stream) {
  (void)in_sizes; (void)n_in; (void)out_size; (void)ws_size;
  const float* X = (const float*)d_in[0];
  const float* bias = (const float*)d_in[1];
  const float* shift = (const float*)d_in[2];
  float* out = (float*)d_out;
  float* ws = (float*)d_ws;

  gr_stageA1<<<dim3(CC * 16), dim3(32), 0, stream>>>(X, ws);
  gr_stageA2<<<dim3(CC), dim3(32), 0, stream>>>(X, ws);
  gr_stageB<<<dim3(BB * CC), dim3(32), 0, stream>>>(X, bias, shift, ws, out);
}